// RGBPoseEstimator_17274358465048
// MI455X (gfx1250) — compile-verified
//
#include <hip/hip_runtime.h>
#include <math.h>

#define IMG_H 1080
#define IMG_W 1920
#define TW 64
#define THT 4
#define TILE_W 66           // TW + 2 halo
#define TILE_H 6            // THT + 2 halo
#define THREADS 256

typedef unsigned int u32x4 __attribute__((ext_vector_type(4)));
typedef int i32x8 __attribute__((ext_vector_type(8)));
typedef int i32x4 __attribute__((ext_vector_type(4)));

// ---------------------------------------------------------------------------
// Kernel A: one thread computes SE3 exp (R, t) + intrinsics into workspace.
// ws layout: [0..8]=R row-major, [9..11]=t, 12=fx,13=fy,14=cx,15=cy,16=1/fx,17=1/fy
// ---------------------------------------------------------------------------
__global__ void se3_setup_kernel(const float* __restrict__ x,
                                 const float* __restrict__ K,
                                 float* __restrict__ ws) {
  float w0 = x[0], w1 = x[1], w2 = x[2];
  float v0 = x[3], v1 = x[4], v2 = x[5];
  float t2 = w0 * w0 + w1 * w1 + w2 * w2;
  bool sm = t2 < 1e-8f;
  float th = sqrtf(sm ? 1.0f : t2);
  float s = sinf(th), c = cosf(th);
  float A = sm ? (1.0f - t2 * (1.0f / 6.0f)) : (s / th);
  float B = sm ? (0.5f - t2 * (1.0f / 24.0f)) : ((1.0f - c) / (th * th));
  float C = sm ? ((1.0f / 6.0f) - t2 * (1.0f / 120.0f)) : ((th - s) / (th * th * th));
  float Wm[3][3] = {{0.f, -w2, w1}, {w2, 0.f, -w0}, {-w1, w0, 0.f}};
  float W2[3][3];
  for (int i = 0; i < 3; ++i)
    for (int j = 0; j < 3; ++j)
      W2[i][j] = Wm[i][0] * Wm[0][j] + Wm[i][1] * Wm[1][j] + Wm[i][2] * Wm[2][j];
  float t[3];
  for (int i = 0; i < 3; ++i) {
    for (int j = 0; j < 3; ++j)
      ws[i * 3 + j] = (i == j ? 1.0f : 0.0f) + A * Wm[i][j] + B * W2[i][j];
    float Vi0 = (i == 0 ? 1.0f : 0.0f) + B * Wm[i][0] + C * W2[i][0];
    float Vi1 = (i == 1 ? 1.0f : 0.0f) + B * Wm[i][1] + C * W2[i][1];
    float Vi2 = (i == 2 ? 1.0f : 0.0f) + B * Wm[i][2] + C * W2[i][2];
    t[i] = Vi0 * v0 + Vi1 * v1 + Vi2 * v2;
  }
  ws[9] = t[0]; ws[10] = t[1]; ws[11] = t[2];
  float fx = K[0], fy = K[4], cx = K[2], cy = K[5];
  ws[12] = fx; ws[13] = fy; ws[14] = cx; ws[15] = cy;
  ws[16] = 1.0f / fx; ws[17] = 1.0f / fy;
}

// ---------------------------------------------------------------------------
// Kernel B: one pixel per thread. 64x4 tile per 256-thread block.
// trg halo tile fetched by the Tensor Data Mover (one DMA per block).
// ---------------------------------------------------------------------------
__global__ void __launch_bounds__(THREADS)
pose_residual_jacobian_kernel(const float* __restrict__ ref_img,
                              const float* __restrict__ ref_depth,
                              const float* __restrict__ trg_img,
                              const float* __restrict__ ws,
                              float* __restrict__ out) {
  __shared__ __align__(16) float tile[TILE_H * TILE_W];   // 6x66 trg window
  __shared__ __align__(16) float obuf[THT][TW * 7];       // output staging

  const int tid = threadIdx.x;
  const int bx = blockIdx.x, by = blockIdx.y;
  const int lc = tid & (TW - 1);
  const int lr = tid >> 6;
  const int c = bx * TW + lc;
  const int r = by * THT + lr;

  // window origin, clamped so the 6x66 tile is fully inside the image
  int wr0 = min(max(by * THT - 1, 0), IMG_H - TILE_H);
  int wc0 = min(max(bx * TW - 1, 0), IMG_W - TILE_W);

  // ---- TDM: one tensor_load_to_lds per block (wave 0 issues) ----
  {
    unsigned lds_base = (unsigned)(size_t)&tile[0];
    unsigned long long gaddr = (unsigned long long)(size_t)trg_img +
                               ((unsigned long long)(wr0 * IMG_W + wc0)) * 4ull;
    u32x4 g0;
    g0.x = 1u;                                   // count=1, user D#, no gather
    g0.y = lds_base;                             // lds_addr
    g0.z = (unsigned)(gaddr & 0xFFFFFFFFu);      // global_addr[31:0]
    g0.w = (unsigned)((gaddr >> 32) & 0x01FFFFFFu) | (2u << 30); // addr[56:32] | type=2
    i32x8 g1;
    g1[0] = 0x00020000;                          // wg_mask=0, data_size=2 (4B), no flags
    g1[1] = (int)((IMG_W & 0xFFFF) << 16);       // tensor_dim0[15:0]
    g1[2] = (int)((IMG_W >> 16) | ((IMG_H & 0xFFFF) << 16));  // dim0 hi | dim1 lo
    g1[3] = (int)((IMG_H >> 16) | (TILE_W << 16));            // dim1 hi | tile_dim0
    g1[4] = TILE_H;                              // tile_dim1 | tile_dim2=0
    g1[5] = IMG_W;                               // tensor_dim0_stride[31:0]
    g1[6] = 0;                                   // stride0 hi | stride1 lo
    g1[7] = 0;                                   // stride1 hi
    i32x4 gz4 = {0, 0, 0, 0};                    // 2D tensor: groups 2/3 unused
    i32x8 gz8 = {0, 0, 0, 0, 0, 0, 0, 0};        // unused trailing group (clang-23 arity)
    if (tid < 32)
      __builtin_amdgcn_tensor_load_to_lds(g0, g1, gz4, gz4, gz8, 0);
  }

  // ---- uniform pose/intrinsics (L0-resident, 18 floats) ----
  float R00 = ws[0], R01 = ws[1], R02 = ws[2];
  float R10 = ws[3], R11 = ws[4], R12 = ws[5];
  float R20 = ws[6], R21 = ws[7], R22 = ws[8];
  float t0 = ws[9], t1 = ws[10], t2 = ws[11];
  float fx = ws[12], fy = ws[13], cx = ws[14], cy = ws[15];
  float rfx = ws[16], rfy = ws[17];

  const int idx = r * IMG_W + c;
  float z = ref_depth[idx];

  // back-project
  float xp = ((float)c - cx) * z * rfx;
  float yp = ((float)r - cy) * z * rfy;

  // rigid transform
  float px = R00 * xp + R01 * yp + R02 * z + t0;
  float py = R10 * xp + R11 * yp + R12 * z + t1;
  float pz = R20 * xp + R21 * yp + R22 * z + t2;

  // project
  float rpz = __builtin_amdgcn_rcpf(pz);
  float uw = fx * px * rpz + cx;
  float vw = fy * py * rpz + cy;
  int ui = (int)uw;   // trunc toward zero == astype(int32)
  int vi = (int)vw;
  bool valid = (vi < IMG_H) && (ui < IMG_W) && (vi > 0) && (ui > 0);
  int fl = min(max(vi, 0), IMG_H - 1) * IMG_W + min(max(ui, 0), IMG_W - 1);
  float warped = ref_img[fl];

  // drain the TDM transfer (wave 0 owns TENSORcnt), publish to all waves
  if (tid < 32)
    __builtin_amdgcn_s_wait_tensorcnt(0);
  __syncthreads();

  // tile coords for this pixel; neighbor indices clamped to the window
  // (clamped values are only consumed where gx/gy are forced to zero)
  int tro = r - wr0;                       // 0..5
  int tco = c - wc0;                       // 0..65
  int trm = max(tro - 1, 0), trp = min(tro + 1, TILE_H - 1);
  int tcm = max(tco - 1, 0), tcp = min(tco + 1, TILE_W - 1);

  float tl = tile[trm * TILE_W + tcm], tt = tile[trm * TILE_W + tco], trr = tile[trm * TILE_W + tcp];
  float ml = tile[tro * TILE_W + tcm], mc = tile[tro * TILE_W + tco], mr  = tile[tro * TILE_W + tcp];
  float bl = tile[trp * TILE_W + tcm], bb = tile[trp * TILE_W + tco], br  = tile[trp * TILE_W + tcp];

  float res = valid ? (warped - mc) : 0.0f;

  // Sobel with 2-pixel zero border (matches crop[1:-1]+pad(2) of VALID conv)
  bool interior = (r >= 2) && (r < IMG_H - 2) && (c >= 2) && (c < IMG_W - 2);
  float gx = interior ? (0.125f * (trr - tl) + 0.25f * (mr - ml) + 0.125f * (br - bl)) : 0.0f;
  float gy = interior ? (0.125f * (bl - tl) + 0.25f * (bb - tt) + 0.125f * (br - trr)) : 0.0f;

  // J = gx * r0 + gy * r1  (warp Jacobian of the ORIGINAL back-projected pts)
  float zi = __builtin_amdgcn_rcpf(z);
  float zi2 = zi * zi;
  float fxzi = fx * zi;
  float a0 = -fx * xp * yp * zi2,          b0 = -fx * (1.0f + yp * yp * zi2);
  float a1 =  fx * (1.0f + xp * xp * zi2), b1 =  fx * xp * yp * zi2;
  float a2 = -yp * fxzi,                   b2 =  xp * fxzi;
  float a5 = -fx * xp * zi2,               b5 = -fx * yp * zi2;
  float J0 = valid ? (gx * a0 + gy * b0) : 0.0f;
  float J1 = valid ? (gx * a1 + gy * b1) : 0.0f;
  float J2 = valid ? (gx * a2 + gy * b2) : 0.0f;
  float J3 = valid ? (gx * fxzi) : 0.0f;
  float J4 = valid ? (gy * fxzi) : 0.0f;   // r1[4] = fx*zi per reference (fx, not fy)
  float J5 = valid ? (gx * a5 + gy * b5) : 0.0f;

  // ---- stage (pixel,7) rows in LDS, emit coalesced 128-bit stores ----
  float* ob = &obuf[lr][lc * 7];
  ob[0] = res; ob[1] = J0; ob[2] = J1; ob[3] = J2; ob[4] = J3; ob[5] = J4; ob[6] = J5;
  __syncthreads();

  // each 64-pixel row segment = 448 floats = 112 float4, 16B-aligned
  const float4* src = (const float4*)&obuf[lr][0];
  float4* dst = (float4*)out + ((size_t)(r * IMG_W + bx * TW) * 7) / 4;
  for (int j = lc; j < (TW * 7) / 4; j += TW)
    dst[j] = src[j];
}

extern "C" void kernel_launch(void* const* d_in, const int* in_sizes, int n_in,
                              void* d_out, int out_size, void* d_ws, size_t ws_size,
                              hipStream_t stream) {
  const float* x          = (const float*)d_in[0];
  const float* ref_img    = (const float*)d_in[1];
  const float* ref_depth  = (const float*)d_in[2];
  const float* trg_img    = (const float*)d_in[3];
  const float* intrinsics = (const float*)d_in[4];
  float* out = (float*)d_out;
  float* ws  = (float*)d_ws;

  se3_setup_kernel<<<1, 1, 0, stream>>>(x, intrinsics, ws);

  dim3 grid(IMG_W / TW, IMG_H / THT);
  pose_residual_jacobian_kernel<<<grid, THREADS, 0, stream>>>(
      ref_img, ref_depth, trg_img, ws, out);
}